// MoE_Buffer_15968688406556
// MI455X (gfx1250) — compile-verified
//
#include <hip/hip_runtime.h>
#include <hip/hip_bf16.h>

typedef float v2f __attribute__((ext_vector_type(2)));
typedef float v4f __attribute__((ext_vector_type(4)));
typedef float v8f __attribute__((ext_vector_type(8)));

// Problem constants (from reference): B=256, H=W=64, D=256, E=8
#define K_TOT   (1u << 20)      // HW*D = 1,048,576 features per row
#define B_ROWS  256
#define N_EXP   8
#define N_GROUP 16              // 256 rows / 16-row WMMA tiles
#define N_SLICE 256             // K-slices for parallelism (K per slice = 4096)
#define KS      4096
#define KT      512             // LDS tile depth per buffer
#define LPAD    (KT + 4)        // LDS row padding to break bank conflicts

// ---------------- workspace layout (floats then ints) ----------------
// wsF[0      .. 4095 ] : dots   [256 rows][16 padded experts]
// wsF[4096   .. 4351 ] : ssx    [256]  (||x_b||^2)
// wsF[4352   .. 4367 ] : ssm    [16]   (||m_e||^2, 8 used)
// wsI = (int*)(wsF + 8192):
//   wsI[0..15]  : offsets[9] (expert segment offsets)
//   wsI[16..271]: order[256] (rows sorted by expert)

// ---- CDNA5 async global->LDS copy (ASYNCcnt-tracked, no VGPR round-trip) ----
__device__ __forceinline__ void async_copy_b128(const void* gptr, void* lptr) {
    // Flat LDS pointers carry the wave-relative LDS offset in addr[31:0].
    unsigned lds_off = (unsigned)(unsigned long long)lptr;
    asm volatile("global_load_async_to_lds_b128 %0, %1, off"
                 :: "v"(lds_off), "v"(gptr) : "memory");
}
__device__ __forceinline__ void async_wait0() {
    asm volatile("s_wait_asynccnt 0" ::: "memory");
}

__global__ __launch_bounds__(256) void k_zero(float* wsF) {
    int i = blockIdx.x * 256 + threadIdx.x;
    if (i < 4368) wsF[i] = 0.0f;
}

// ---- kernel 1: dots[b,e] via V_WMMA_F32_16X16X4_F32, double-buffered async LDS ----
__global__ __launch_bounds__(256) void k_dots(const float* __restrict__ x,
                                              const float* __restrict__ mem,
                                              float* __restrict__ gdots,
                                              float* __restrict__ gssx) {
    __shared__ float lx[2][16][LPAD];   // x tiles (double buffered)
    __shared__ float lm[2][8][LPAD];    // memory tiles (double buffered)
    __shared__ float red[256];          // 16x16 dot reduction across waves
    __shared__ float ssq[16];           // per-row sumsq reduction

    const int tid  = threadIdx.x;
    const int g    = blockIdx.x & (N_GROUP - 1);   // row group
    const int s    = blockIdx.x >> 4;              // K slice
    const int lane = tid & 31;
    const int wave = tid >> 5;

    red[tid] = 0.0f;
    if (tid < 16) ssq[tid] = 0.0f;

    const size_t k0base = (size_t)s * KS;

    // Issue one chunk's async loads (each thread: 8 x-tiles + 4 m-tiles of b128)
    auto issue = [&](int kt, int buf) {
        #pragma unroll
        for (int it = 0; it < 8; ++it) {            // 16*KT floats of x
            int f  = (it * 256 + tid) * 4;
            int r  = f >> 9;                        // /KT
            int kk = f & (KT - 1);
            async_copy_b128(x + (((size_t)(g * 16 + r)) << 20) + k0base + kt + kk,
                            &lx[buf][r][kk]);
        }
        #pragma unroll
        for (int it = 0; it < 4; ++it) {            // 8*KT floats of memory
            int f  = (it * 256 + tid) * 4;
            int r  = f >> 9;
            int kk = f & (KT - 1);
            async_copy_b128(mem + (((size_t)r) << 20) + k0base + kt + kk,
                            &lm[buf][r][kk]);
        }
    };

    v8f  c = {};                        // 16x16 f32 accumulator (8 VGPRs/lane)
    float ssa[4] = {0.f, 0.f, 0.f, 0.f}; // sum(x^2), 4 rotating accumulators

    const int row = lane & 15;          // A-matrix M (and B-matrix N) index
    const int kh2 = (lane >> 4) << 1;   // lanes 0-15: K0,K1 ; lanes 16-31: K2,K3

    issue(0, 0);
    for (int kt = 0; kt < KS; kt += KT) {
        const int buf = (kt >> 9) & 1;
        async_wait0();       // my loads into buf landed in LDS
        __syncthreads();     // everyone's loads landed; prev compute on buf^1 done
        if (kt + KT < KS) issue(kt + KT, buf ^ 1);   // overlap with WMMA below

        #pragma unroll
        for (int t = 0; t < 16; ++t) {
            const int kk = wave * 64 + t * 4 + kh2;
            v2f a;                                   // A: x tile 16x4 (MxK)
            a.x = lx[buf][row][kk];
            a.y = lx[buf][row][kk + 1];
            v2f b;                                   // B: mem^T tile 4x16 (KxN)
            if (row < N_EXP) { b.x = lm[buf][row][kk]; b.y = lm[buf][row][kk + 1]; }
            else             { b.x = 0.0f;             b.y = 0.0f;                 }
            c = __builtin_amdgcn_wmma_f32_16x16x4_f32(false, a, false, b,
                                                      (short)0, c, false, false);
            ssa[t & 3] = __builtin_fmaf(a.x, a.x, ssa[t & 3]);
            ssa[t & 3] = __builtin_fmaf(a.y, a.y, ssa[t & 3]);
        }
    }
    const float ss = (ssa[0] + ssa[1]) + (ssa[2] + ssa[3]);

    // Cross-wave reduction in LDS, then global f32 atomics into ws accumulators.
    // C/D layout: VGPR r, lanes 0-15 -> M=r, N=lane ; lanes 16-31 -> M=r+8, N=lane-16
    const int mbase = (lane >> 4) << 3;
    #pragma unroll
    for (int r = 0; r < 8; ++r)
        atomicAdd(&red[(r + mbase) * 16 + (lane & 15)], c[r]);
    atomicAdd(&ssq[lane & 15], ss);
    __syncthreads();

    if (tid < 16) atomicAdd(&gssx[g * 16 + tid], ssq[tid]);
    atomicAdd(&gdots[(g * 16 + (tid >> 4)) * 16 + (tid & 15)], red[tid]);
}

// ---- kernel 2: per-expert ||m||^2 (one block per expert) ----
__global__ __launch_bounds__(256) void k_mnorm(const float* __restrict__ mem,
                                               float* __restrict__ gssm) {
    __shared__ float r[256];
    const int tid = threadIdx.x;
    const int e   = blockIdx.x;
    float ss = 0.0f;
    for (unsigned i = tid * 4; i < K_TOT; i += 256 * 4) {
        v4f v = *(const v4f*)(mem + (((size_t)e) << 20) + i);
        ss = __builtin_fmaf(v.x, v.x, ss);
        ss = __builtin_fmaf(v.y, v.y, ss);
        ss = __builtin_fmaf(v.z, v.z, ss);
        ss = __builtin_fmaf(v.w, v.w, ss);
    }
    r[tid] = ss;
    __syncthreads();
    for (int o = 128; o > 0; o >>= 1) {
        if (tid < o) r[tid] += r[tid + o];
        __syncthreads();
    }
    if (tid == 0) gssm[e] = r[0];
}

// ---- kernel 3: cosine argmax + build expert-sorted row permutation ----
__global__ __launch_bounds__(256) void k_argmax(const float* __restrict__ gdots,
                                                const float* __restrict__ gssx,
                                                const float* __restrict__ gssm,
                                                float* __restrict__ outExp,
                                                int* __restrict__ gOff,
                                                int* __restrict__ gOrder) {
    __shared__ int sExp[B_ROWS];
    const int b = threadIdx.x;
    const float nx = sqrtf(gssx[b]);
    float best = -3.402823e38f;
    int   be   = 0;
    #pragma unroll
    for (int e = 0; e < N_EXP; ++e) {
        float denom = fmaxf(nx * sqrtf(gssm[e]), 1e-8f);
        float sim   = gdots[b * 16 + e] / denom;
        if (sim > best) { best = sim; be = e; }     // first-max tie rule (argmax)
    }
    sExp[b]   = be;
    outExp[b] = (float)be;
    __syncthreads();
    if (b == 0) {
        int cnt[N_EXP];
        #pragma unroll
        for (int e = 0; e < N_EXP; ++e) cnt[e] = 0;
        for (int i = 0; i < B_ROWS; ++i) cnt[sExp[i]]++;
        int off[N_EXP + 1]; off[0] = 0;
        #pragma unroll
        for (int e = 0; e < N_EXP; ++e) off[e + 1] = off[e] + cnt[e];
        for (int e = 0; e <= N_EXP; ++e) gOff[e] = off[e];
        int cur[N_EXP];
        #pragma unroll
        for (int e = 0; e < N_EXP; ++e) cur[e] = off[e];
        for (int i = 0; i < B_ROWS; ++i) gOrder[cur[sExp[i]]++] = i;
    }
}

// ---- kernel 4: fused segment-mean + memory EMA update ----
__global__ __launch_bounds__(256) void k_update(const float* __restrict__ x,
                                                const float* __restrict__ mem,
                                                const int* __restrict__ gOff,
                                                const int* __restrict__ gOrder,
                                                float* __restrict__ outMem) {
    __shared__ int so[B_ROWS];
    __shared__ int soff[N_EXP + 1];
    const int tid = threadIdx.x;
    so[tid] = gOrder[tid];
    if (tid <= N_EXP) soff[tid] = gOff[tid];
    __syncthreads();

    const size_t i = ((size_t)blockIdx.x * 256 + tid) * 4;   // float4 column
    #pragma unroll 1
    for (int e = 0; e < N_EXP; ++e) {
        const int beg = soff[e], end = soff[e + 1];
        const int cnt = end - beg;
        v4f acc = {0.0f, 0.0f, 0.0f, 0.0f};
        for (int j = beg; j < end; ++j) {
            const int brow = so[j];   // wave-uniform
            if (j + 1 < end) {        // one-row-ahead prefetch (global_prefetch_b8)
                __builtin_prefetch(x + (((size_t)so[j + 1]) << 20) + i, 0, 0);
            }
            v4f v = __builtin_nontemporal_load(
                        (const v4f*)(x + (((size_t)brow) << 20) + i));
            acc += v;
        }
        v4f mv = *(const v4f*)(mem + (((size_t)e) << 20) + i);
        v4f o;
        if (cnt > 0) {
            const float inv = 1.0f / (float)cnt;
            o = (mv + acc * inv) * 0.5f;
        } else {
            o = mv;
        }
        *(v4f*)(outMem + (((size_t)e) << 20) + i) = o;
    }
}

extern "C" void kernel_launch(void* const* d_in, const int* in_sizes, int n_in,
                              void* d_out, int out_size, void* d_ws, size_t ws_size,
                              hipStream_t stream) {
    const float* x   = (const float*)d_in[0];   // [256, 64, 64, 256]
    const float* mem = (const float*)d_in[1];   // [8, 4096, 256]

    float* wsF   = (float*)d_ws;
    float* dots  = wsF;               // 4096
    float* ssx   = wsF + 4096;        // 256
    float* ssm   = wsF + 4352;        // 16
    int*   wsI   = (int*)(wsF + 8192);
    int*   gOff  = wsI;               // 9 (padded 16)
    int*   gOrd  = wsI + 16;          // 256

    float* outExp = (float*)d_out;        // [256] expert index as float
    float* outMem = outExp + B_ROWS;      // [8, 4096, 256]

    k_zero  <<<18, 256, 0, stream>>>(wsF);
    k_dots  <<<N_GROUP * N_SLICE, 256, 0, stream>>>(x, mem, dots, ssx);
    k_mnorm <<<N_EXP, 256, 0, stream>>>(mem, ssm);
    k_argmax<<<1, 256, 0, stream>>>(dots, ssx, ssm, outExp, gOff, gOrd);
    k_update<<<(K_TOT / 4) / 256, 256, 0, stream>>>(x, mem, gOff, gOrd, outMem);
}